// GeoGraph_16741782520369
// MI455X (gfx1250) — compile-verified
//
#include <hip/hip_runtime.h>
#include <hip/hip_bf16.h>
#include <math.h>

#define N_POI   50000
#define DIM     128
#define E_EDGES 400000
#define GCN_NUM 3
#define HEADS   8
#define BATCH   64
#define SEQ     100
#define LEAKY   0.01f

typedef __attribute__((ext_vector_type(2))) float v2f;
typedef __attribute__((ext_vector_type(8))) float v8f;

#define CEILDIV(a,b) (((a)+(b)-1)/(b))

// ---------------- graph preprocessing ----------------

__global__ void k_init_deg(float* deg) {
  int i = blockIdx.x * blockDim.x + threadIdx.x;
  if (i < N_POI) deg[i] = 1.0f;  // self-loop contributes 1 to every degree
}

__global__ void k_count_deg(const int* __restrict__ edges, float* deg) {
  int e = blockIdx.x * blockDim.x + threadIdx.x;
  if (e < E_EDGES) {
    // symmetrized graph: edge counted on both endpoints
    atomicAdd(&deg[edges[e]], 1.0f);
    atomicAdd(&deg[edges[E_EDGES + e]], 1.0f);
  }
}

__global__ void k_rsqrt_deg(float* deg) {
  int i = blockIdx.x * blockDim.x + threadIdx.x;
  if (i < N_POI) deg[i] = rsqrtf(deg[i]);   // deg >= 1 always
}

__global__ void k_edge_w(const float* __restrict__ dv, const int* __restrict__ edges,
                         const float* __restrict__ dis, float* __restrict__ w) {
  int e = blockIdx.x * blockDim.x + threadIdx.x;
  if (e < E_EDGES) {
    float d = dv[e];
    w[e] = expf(-d * d) * dis[edges[e]] * dis[edges[E_EDGES + e]];
  }
}

// agg[i] = dis[i]^2 * enc[i]   (self-loop term; also initializes agg)
__global__ void k_self_init(const float* __restrict__ enc, const float* __restrict__ dis,
                            float* __restrict__ agg) {
  int i = blockIdx.x * blockDim.x + threadIdx.x;
  if (i < N_POI * DIM) {
    float s = dis[i >> 7];
    agg[i] = s * s * enc[i];
  }
}

// One wave per directed edge; lane covers 4 consecutive floats of the 128-wide row.
__global__ void k_aggregate(const float* __restrict__ enc, const int* __restrict__ edges,
                            const float* __restrict__ w, float* __restrict__ agg) {
  int gt   = blockIdx.x * blockDim.x + threadIdx.x;
  int wid  = gt >> 5;
  int lane = gt & 31;
  if (wid >= 2 * E_EDGES) return;              // wave-uniform
  int e = wid >> 1;
  int a = edges[e];
  int b = edges[E_EDGES + e];
  if (wid & 1) { int t = a; a = b; b = t; }
  float we = w[e];
  const float4 s4 = *(const float4*)(enc + (size_t)b * DIM + lane * 4);
  float* dst = agg + (size_t)a * DIM + lane * 4;
  atomicAdd(dst + 0, we * s4.x);
  atomicAdd(dst + 1, we * s4.y);
  atomicAdd(dst + 2, we * s4.z);
  atomicAdd(dst + 3, we * s4.w);
}

// ---------------- fused GCN dense layer: out = l2norm(leaky(agg @ W^T + b)) ----------------
// grid.x = N_POI/16 tiles of 16 rows; 256 threads = 8 waves, wave j computes cols [16j,16j+16)
__global__ void __launch_bounds__(256)
k_gcn_layer(const float* __restrict__ agg, const float* __restrict__ W,
            const float* __restrict__ bias, float* __restrict__ out) {
  __shared__ float sA[16 * DIM];
  __shared__ float sO[16 * DIM];
  __shared__ float sInv[16];
  const int tid   = threadIdx.x;
  const int wave  = tid >> 5;
  const int lane  = tid & 31;
  const int hf    = lane >> 4;       // which half of the wave
  const int mrow  = lane & 15;
  const int tileM = blockIdx.x * 16;
  const int jt    = wave * 16;       // output column tile

  // stage 16x128 A tile in LDS (8 floats per thread)
  for (int i = tid; i < 16 * DIM; i += 256)
    sA[i] = agg[(size_t)(tileM + (i >> 7)) * DIM + (i & 127)];
  __syncthreads();

  v8f acc = {};
#pragma unroll
  for (int k0 = 0; k0 < DIM; k0 += 4) {
    v2f a, b;
    // A 16x4 f32 layout: lanes 0-15 hold K={0,1}, lanes 16-31 hold K={2,3}
    a.x = sA[mrow * DIM + k0 + 2 * hf + 0];
    a.y = sA[mrow * DIM + k0 + 2 * hf + 1];
    // B[k][n] = W[jt+n][k]  (enc = agg @ W^T)
    b.x = W[(size_t)(jt + mrow) * DIM + k0 + 2 * hf + 0];
    b.y = W[(size_t)(jt + mrow) * DIM + k0 + 2 * hf + 1];
    acc = __builtin_amdgcn_wmma_f32_16x16x4_f32(false, a, false, b, (short)0, acc,
                                                false, false);
  }

  // D layout: VGPR v -> row v + 8*hf, col = mrow. Bias + leaky-ReLU into LDS.
  float bj = bias[jt + mrow];
#pragma unroll
  for (int v = 0; v < 8; ++v) {
    int r = v + 8 * hf;
    float x = acc[v] + bj;
    x = x > 0.0f ? x : LEAKY * x;
    sO[r * DIM + jt + mrow] = x;
  }
  __syncthreads();

  if (tid < 16) {
    float s = 0.0f;
    for (int c = 0; c < DIM; ++c) { float x = sO[tid * DIM + c]; s += x * x; }
    sInv[tid] = 1.0f / fmaxf(sqrtf(s), 1e-12f);
  }
  __syncthreads();

  for (int i = tid; i < 16 * DIM; i += 256) {
    int r = i >> 7;
    out[(size_t)(tileM + r) * DIM + (i & 127)] = sO[i] * sInv[r];
  }
}

// ---------------- generic WMMA GEMM (K=128): C[m,n] = sum_k A[m,k]*Wt[n,k] + bias[n] ----
// one wave per 16x16 tile; M and N multiples of 16; 128 threads (4 waves) per block
__global__ void __launch_bounds__(128)
k_gemm_bias(const float* __restrict__ A, const float* __restrict__ Wt,
            const float* __restrict__ bias, float* __restrict__ C, int M, int N) {
  int wg   = (blockIdx.x * blockDim.x + threadIdx.x) >> 5;
  int lane = threadIdx.x & 31;
  int ntn  = N >> 4;
  int tiles = (M >> 4) * ntn;
  if (wg >= tiles) return;                      // wave-uniform
  int tm = (wg / ntn) << 4;
  int tn = (wg % ntn) << 4;
  int hf = lane >> 4, mrow = lane & 15;

  v8f acc = {};
#pragma unroll
  for (int k0 = 0; k0 < DIM; k0 += 4) {
    v2f a, b;
    a.x = A[(size_t)(tm + mrow) * DIM + k0 + 2 * hf + 0];
    a.y = A[(size_t)(tm + mrow) * DIM + k0 + 2 * hf + 1];
    b.x = Wt[(size_t)(tn + mrow) * DIM + k0 + 2 * hf + 0];
    b.y = Wt[(size_t)(tn + mrow) * DIM + k0 + 2 * hf + 1];
    acc = __builtin_amdgcn_wmma_f32_16x16x4_f32(false, a, false, b, (short)0, acc,
                                                false, false);
  }
  float bj = bias[tn + mrow];
#pragma unroll
  for (int v = 0; v < 8; ++v)
    C[(size_t)(tm + v + 8 * hf) * N + tn + mrow] = acc[v] + bj;
}

// ---------------- gathers / attention / mean ----------------

__global__ void k_gather(const float* __restrict__ enc, const int* __restrict__ idx,
                         float* __restrict__ out, int n_elems) {
  int i = blockIdx.x * blockDim.x + threadIdx.x;
  if (i < n_elems) out[i] = enc[(size_t)idx[i >> 7] * DIM + (i & 127)];
}

// one block per (batch, head): scalar softmax attention, q/k/v/scores in LDS
__global__ void __launch_bounds__(256)
k_attn(const float* __restrict__ qkv, float* __restrict__ o) {
  int b = blockIdx.x >> 3;
  int h = blockIdx.x & 7;
  __shared__ float sq[SEQ * 16], sk[SEQ * 16], sv[SEQ * 16];
  __shared__ float sS[SEQ * SEQ];
  int tid = threadIdx.x;

  for (int i = tid; i < SEQ * 16; i += 256) {
    int l = i >> 4, d = i & 15;
    size_t base = (size_t)(b * SEQ + l) * (3 * DIM) + h * 16 + d;
    sq[i] = qkv[base];
    sk[i] = qkv[base + DIM];
    sv[i] = qkv[base + 2 * DIM];
  }
  __syncthreads();

  for (int i = tid; i < SEQ * SEQ; i += 256) {
    int r = i / SEQ, c = i % SEQ;
    float s = 0.0f;
#pragma unroll
    for (int d = 0; d < 16; ++d) s += sq[r * 16 + d] * sk[c * 16 + d];
    sS[i] = s * 0.25f;   // 1/sqrt(16)
  }
  __syncthreads();

  if (tid < SEQ) {
    float mx = -INFINITY;
    for (int c = 0; c < SEQ; ++c) mx = fmaxf(mx, sS[tid * SEQ + c]);
    float sum = 0.0f;
    for (int c = 0; c < SEQ; ++c) { float e = expf(sS[tid * SEQ + c] - mx); sS[tid * SEQ + c] = e; sum += e; }
    float inv = 1.0f / sum;
    for (int c = 0; c < SEQ; ++c) sS[tid * SEQ + c] *= inv;
  }
  __syncthreads();

  for (int i = tid; i < SEQ * 16; i += 256) {
    int l = i >> 4, d = i & 15;
    float s = 0.0f;
    for (int j = 0; j < SEQ; ++j) s += sS[l * SEQ + j] * sv[j * 16 + d];
    o[(size_t)(b * SEQ + l) * DIM + h * 16 + d] = s;
  }
}

__global__ void k_mean(const float* __restrict__ o, float* __restrict__ om) {
  int i = blockIdx.x * blockDim.x + threadIdx.x;
  if (i < BATCH * DIM) {
    int b = i >> 7, c = i & 127;
    float s = 0.0f;
    for (int l = 0; l < SEQ; ++l) s += o[(size_t)(b * SEQ + l) * DIM + c];
    om[i] = s * (1.0f / SEQ);
  }
}

// ---------------- launch ----------------

extern "C" void kernel_launch(void* const* d_in, const int* in_sizes, int n_in,
                              void* d_out, int out_size, void* d_ws, size_t ws_size,
                              hipStream_t stream) {
  const float* embeds     = (const float*)d_in[0];
  const float* gcn_W      = (const float*)d_in[1];
  const float* gcn_b      = (const float*)d_in[2];
  const float* in_proj_w  = (const float*)d_in[3];
  const float* in_proj_b  = (const float*)d_in[4];
  const float* out_proj_w = (const float*)d_in[5];
  const float* out_proj_b = (const float*)d_in[6];
  const float* dist_vec   = (const float*)d_in[7];
  const int*   dist_edges = (const int*)d_in[8];
  const int*   x_idx      = (const int*)d_in[9];
  const int*   poi_idx    = (const int*)d_in[10];
  float* out = (float*)d_out;   // [0,8192): aggr_feat, [8192,16384): tar_embed

  float* ws   = (float*)d_ws;
  float* dis  = ws;                         // 50048 (deg then rsqrt in place)
  float* wbuf = dis + 50048;                // 400000
  float* P0   = wbuf + E_EDGES;             // 6.4M  (agg; later geo+qkv)
  float* P1   = P0 + (size_t)N_POI * DIM;   // 6.4M  (enc ping; final enc)
  float* P2   = P1 + (size_t)N_POI * DIM;   // 6.4M  (enc pong; later o+omean)
  float* geo  = P0;
  float* qkv  = P0 + (size_t)BATCH * SEQ * DIM;
  float* obuf = P2;
  float* omean = P2 + (size_t)BATCH * SEQ * DIM;

  const int ND = N_POI * DIM;

  // degrees -> dis -> edge weights
  k_init_deg<<<CEILDIV(N_POI, 256), 256, 0, stream>>>(dis);
  k_count_deg<<<CEILDIV(E_EDGES, 256), 256, 0, stream>>>(dist_edges, dis);
  k_rsqrt_deg<<<CEILDIV(N_POI, 256), 256, 0, stream>>>(dis);
  k_edge_w<<<CEILDIV(E_EDGES, 256), 256, 0, stream>>>(dist_vec, dist_edges, dis, wbuf);

  // 3 GCN layers (ping-pong: embeds->P1->P2->P1)
  const float* enc_in = embeds;
  float* enc_out = P1;
  for (int layer = 0; layer < GCN_NUM; ++layer) {
    k_self_init<<<CEILDIV(ND, 256), 256, 0, stream>>>(enc_in, dis, P0);
    k_aggregate<<<CEILDIV(2 * E_EDGES * 32, 256), 256, 0, stream>>>(enc_in, dist_edges, wbuf, P0);
    k_gcn_layer<<<N_POI / 16, 256, 0, stream>>>(P0, gcn_W + (size_t)layer * DIM * DIM,
                                                gcn_b + (size_t)layer * DIM, enc_out);
    enc_in  = enc_out;
    enc_out = (enc_out == P1) ? P2 : P1;
  }
  const float* enc = enc_in;   // == P1

  // gathers: session features + target embeds (second half of d_out)
  k_gather<<<CEILDIV(BATCH * SEQ * DIM, 256), 256, 0, stream>>>(enc, x_idx, geo, BATCH * SEQ * DIM);
  k_gather<<<CEILDIV(BATCH * DIM, 256), 256, 0, stream>>>(enc, poi_idx, out + BATCH * DIM, BATCH * DIM);

  // QKV projection: [6400,128] x [384,128]^T
  {
    int tiles = (BATCH * SEQ / 16) * (3 * DIM / 16);
    k_gemm_bias<<<CEILDIV(tiles * 32, 128), 128, 0, stream>>>(geo, in_proj_w, in_proj_b,
                                                              qkv, BATCH * SEQ, 3 * DIM);
  }

  // attention per (batch, head)
  k_attn<<<BATCH * HEADS, 256, 0, stream>>>(qkv, obuf);

  // mean over L, then out-projection (linear commutes with mean)
  k_mean<<<CEILDIV(BATCH * DIM, 256), 256, 0, stream>>>(obuf, omean);
  {
    int tiles = (BATCH / 16) * (DIM / 16);
    k_gemm_bias<<<CEILDIV(tiles * 32, 128), 128, 0, stream>>>(omean, out_proj_w, out_proj_b,
                                                              out, BATCH, DIM);
  }
}